// AdaptiveRankFusionLayer_40673340293781
// MI455X (gfx1250) — compile-verified
//
#include <hip/hip_runtime.h>

// CP low-rank fusion for MI455X (gfx1250, wave32).
// out[B,512] = ((x1@f1) * (x2@f2) * (x3@f3)) @ f_out.T, RANK=10 padded to 16.
// Memory-bound: ~740MB @ 23.3TB/s => ~32us floor. Projections run on
// V_WMMA_F32_16X16X4_F32 (exact fp32). x tiles staged through LDS; next chunk
// is warmed with global_prefetch_b8 (zero VGPR cost) instead of a register
// double-buffer (which spilled to scratch in the previous revision).

typedef __attribute__((ext_vector_type(2))) float v2f;
typedef __attribute__((ext_vector_type(8))) float v8f;

#define B_TOTAL        65536
#define K1             1024
#define K2             512
#define K3             768
#define OUT_N          512
#define RANK           10
#define KC             64      // K-chunk staged in LDS per wave
#define XSTR           68      // padded LDS row stride (floats) -> conflict-free ds_load_b64
#define ROWS_PER_WAVE  16
#define NWAVES         8
#define ROWS_PER_BLOCK (ROWS_PER_WAVE * NWAVES)   // 128

// One wave computes P = x_tile[16,K] @ f[K,16].
// A 16x4 f32 frag: VGPR j, lane L -> A[M = L%16][K = j + 2*(L>>4)]
// B 4x16 f32 frag: VGPR j, lane L -> B[K = j + 2*(L>>4)][N = L%16]
// C/D 16x16 f32:   VGPR j, lane L -> D[M = j + 8*(L>>4)][N = L%16]
// Columns N in [RANK,16) of B/D are junk by construction (clamped loads);
// stage 2 only reads y[:,0:RANK], so no masking / exec divergence needed.
__device__ __forceinline__ v8f project16(const float* __restrict__ x,
                                         const float* __restrict__ f,
                                         int K, int m0,
                                         float (*__restrict__ xt)[XSTR],
                                         int lane, int half, int lrow)
{
    const int lrow_c = (lrow < RANK) ? lrow : (RANK - 1);  // in-bounds clamp only

    // Staging geometry: slot i covers rows {2i + half} of the 16xKC tile,
    // cols (lane&15)*4 .. +3. Single base pointer; slot stride = 2*K floats
    // folds into 24-bit immediate offsets (max 7*8*K bytes < 8M).
    const float* gsrc = x + (size_t)(m0 + half) * K + ((lane & 15) << 2);
    float*       ldst = &xt[half][(lane & 15) << 2];       // slot stride 2*XSTR

    v8f acc = {};
    for (int kk0 = 0; kk0 < K; kk0 += KC) {
        // ---- stage 16 x KC tile: coalesced b128 loads -> ds_store_b128 ----
        #pragma unroll
        for (int i = 0; i < 8; ++i) {
            float4 v = *(const float4*)(gsrc + (size_t)(2 * i) * K);
            *(float4*)(ldst + 2 * i * XSTR) = v;
        }
        // ---- warm the next chunk: global_prefetch_b8, no VGPRs, no LOADcnt.
        // Speculative prefetch past the tensor end is silently dropped.
        #pragma unroll
        for (int i = 0; i < 8; ++i)
            __builtin_prefetch(gsrc + (size_t)(2 * i) * K + KC, 0, 0);

        // ---- 16 WMMAs over the LDS-resident chunk ----
        const float* fch = f + (size_t)(kk0 + 2 * half) * RANK + lrow_c;
        #pragma unroll
        for (int k = 0; k < KC; k += 4) {
            v2f a = *(const v2f*)&xt[lrow][k + 2 * half];   // ds_load_b64
            v2f b;
            b.x = fch[k * RANK];           // unconditional, L2/WGP$-resident
            b.y = fch[k * RANK + RANK];
            acc = __builtin_amdgcn_wmma_f32_16x16x4_f32(
                      false, a, false, b, (short)0, acc, false, false);
        }
        gsrc += KC;
    }
    return acc;
}

__global__ __launch_bounds__(256)
void cp_fusion_kernel(const float* __restrict__ x1, const float* __restrict__ x2,
                      const float* __restrict__ x3, const float* __restrict__ f1,
                      const float* __restrict__ f2, const float* __restrict__ f3,
                      const float* __restrict__ f_out, float* __restrict__ out)
{
    __shared__ float lds_x[NWAVES][ROWS_PER_WAVE][XSTR];   // ~34 KB
    __shared__ float y_lds[ROWS_PER_BLOCK][16];            //   8 KB

    const int tid  = threadIdx.x;
    const int lane = tid & 31;
    const int wave = tid >> 5;
    const int half = lane >> 4;
    const int lrow = lane & 15;

    const int row_base = blockIdx.x * ROWS_PER_BLOCK;
    const int m0       = row_base + wave * ROWS_PER_WAVE;

    // ---- stage 1: three rank-16 projections on the WMMA pipe ----
    v8f p1 = project16(x1, f1, K1, m0, lds_x[wave], lane, half, lrow);
    v8f p2 = project16(x2, f2, K2, m0, lds_x[wave], lane, half, lrow);
    v8f p3 = project16(x3, f3, K3, m0, lds_x[wave], lane, half, lrow);

    // Hadamard product directly in the shared C/D layout, spill to LDS.
    #pragma unroll
    for (int j = 0; j < 8; ++j) {
        float y = p1[j] * p2[j] * p3[j];
        y_lds[wave * ROWS_PER_WAVE + j + 8 * half][lrow] = y;
    }
    __syncthreads();

    // ---- stage 2: out[128,512] = y[128,0:10] @ f_out.T (VALU, tiny) ----
    // Each thread owns 2 output columns; its f_out rows live in registers.
    const int n0 = tid * 2;
    float fr0[RANK], fr1[RANK];
    #pragma unroll
    for (int r = 0; r < RANK; ++r) {
        fr0[r] = f_out[(size_t)n0 * RANK + r];
        fr1[r] = f_out[(size_t)(n0 + 1) * RANK + r];
    }
    for (int m = 0; m < ROWS_PER_BLOCK; ++m) {
        float s0 = 0.0f, s1 = 0.0f;
        #pragma unroll
        for (int r = 0; r < RANK; ++r) {
            float yv = y_lds[m][r];       // broadcast read, conflict-free
            s0 += yv * fr0[r];
            s1 += yv * fr1[r];
        }
        float2 o; o.x = s0; o.y = s1;
        *(float2*)&out[(size_t)(row_base + m) * OUT_N + n0] = o;   // coalesced b64
    }
}

extern "C" void kernel_launch(void* const* d_in, const int* in_sizes, int n_in,
                              void* d_out, int out_size, void* d_ws, size_t ws_size,
                              hipStream_t stream) {
    (void)in_sizes; (void)n_in; (void)out_size; (void)d_ws; (void)ws_size;
    const float* x1 = (const float*)d_in[0];
    const float* x2 = (const float*)d_in[1];
    const float* x3 = (const float*)d_in[2];
    const float* f1 = (const float*)d_in[3];
    const float* f2 = (const float*)d_in[4];
    const float* f3 = (const float*)d_in[5];
    const float* fo = (const float*)d_in[6];
    float* out = (float*)d_out;

    dim3 grid(B_TOTAL / ROWS_PER_BLOCK);   // 512 blocks
    dim3 block(256);                       // 8 wave32s
    hipLaunchKernelGGL(cp_fusion_kernel, grid, block, 0, stream,
                       x1, x2, x3, f1, f2, f3, fo, out);
}